// Model_25881472925698
// MI455X (gfx1250) — compile-verified
//
#include <hip/hip_runtime.h>
#include <hip/hip_bf16.h>
#include <math.h>

typedef __attribute__((ext_vector_type(2))) float v2f;
typedef __attribute__((ext_vector_type(8))) float v8f;
typedef __attribute__((ext_vector_type(4))) unsigned int u32x4;
typedef __attribute__((ext_vector_type(4))) int i32x4;
typedef __attribute__((ext_vector_type(8))) int i32x8;

#define NU 50000
#define NB 50000
#define NE 300000
#define H  256
#define ED 64

#define G_EDGE 6   // row tiles per block, edge kernel  (18750 / 6 = 3125 blocks)
#define G_NODE 5   // row tiles per block, node kernel  (3125 / 5 = 625 blocks)
#define KC     32  // K-chunk rows staged in LDS, node kernel (32*256*4B = 32 KB)

// ---------------------------------------------------------------------------
// LDS byte offset of a __shared__ object: low 32 bits of its generic address.
// ---------------------------------------------------------------------------
__device__ __forceinline__ unsigned lds_off(const void* p) {
  return (unsigned)(unsigned long long)p;
}

// ---------------------------------------------------------------------------
// TDM: DMA `nelem` f32 from global memory into LDS at byte offset `ldsb`.
// 1-D tile descriptor per CDNA5 ISA:
//   group0: count=1, lds_addr, global_addr[56:0], type=2
//   group1: data_size=4B, tensor_dim0 = tile_dim0 = nelem, dims1+ = 1/unused
// Issue from ONE wave only; completion via s_wait_tensorcnt.
// ---------------------------------------------------------------------------
__device__ __forceinline__ void tdm_load_f32(unsigned ldsb, const void* gptr,
                                             unsigned nelem) {
  unsigned long long ga = (unsigned long long)gptr;
  u32x4 g0;
  g0[0] = 1u;                                   // count=1 (valid descriptor)
  g0[1] = ldsb;                                 // lds_addr (bytes)
  g0[2] = (unsigned)(ga & 0xFFFFFFFFu);         // global_addr[31:0]
  g0[3] = (unsigned)((ga >> 32) & 0x01FFFFFFu)  // global_addr[56:32]
          | 0x80000000u;                        // type=2 ("image") in [127:126]
  i32x8 g1;
  g1[0] = (int)(2u << 16);                      // data_size = 2 -> 4 bytes
  g1[1] = (int)((nelem & 0xFFFFu) << 16);       // tensor_dim0[15:0]
  g1[2] = (int)((nelem >> 16) | (1u << 16));    // tensor_dim0[31:16], tensor_dim1=1
  g1[3] = (int)((nelem & 0xFFFFu) << 16);       // tile_dim0 = nelem
  g1[4] = 1;                                    // tile_dim1 = 1, tile_dim2 = 0
  g1[5] = (int)nelem;                           // tensor_dim0_stride[31:0]
  g1[6] = 0;
  g1[7] = 0;
  i32x4 gz = {0, 0, 0, 0};
#if defined(__clang_major__) && (__clang_major__ >= 23)
  i32x8 gz8 = {0, 0, 0, 0, 0, 0, 0, 0};
  __builtin_amdgcn_tensor_load_to_lds(g0, g1, gz, gz, gz8, 0);
#else
  __builtin_amdgcn_tensor_load_to_lds(g0, g1, gz, gz, 0);
#endif
}

// ---------------------------------------------------------------------------
// x[i][h] = emb[nid[i]][h]
// ---------------------------------------------------------------------------
__global__ __launch_bounds__(256) void gather_rows_kernel(
    const float* __restrict__ emb, const int* __restrict__ nid,
    float* __restrict__ x) {
  const int i = blockIdx.x;
  const int h = threadIdx.x;
  x[(size_t)i * H + h] = emb[(size_t)nid[i] * H + h];
}

// ---------------------------------------------------------------------------
__global__ __launch_bounds__(256) void zero_f32_kernel(float* __restrict__ p,
                                                       size_t n) {
  size_t i = (size_t)blockIdx.x * blockDim.x + threadIdx.x;
  if (i < n) p[i] = 0.0f;
}

// ---------------------------------------------------------------------------
// Fused GINE message + aggregate:
//   m = relu(x_src[src[e]] + edge_attr[e] @ W_edge + b_edge); atomicAdd(agg[dst[e]], m)
// Block = 128 threads (4 waves); W_edge (64 KB) TDM-staged in LDS once, then
// G_EDGE row tiles of 16 edges each. Wave w owns cols [64w,64w+64) as 4 WMMA
// f32 16x16x4 column tiles; B fragments come from LDS.
// ---------------------------------------------------------------------------
__global__ __launch_bounds__(128) void gine_edge_kernel(
    const float* __restrict__ edge_attr,   // [E, 64]
    const int* __restrict__ edge_index,    // [2, E]
    const float* __restrict__ x_src,       // [Nsrc, 256]
    const float* __restrict__ W_edge,      // [64, 256]
    const float* __restrict__ b_edge,      // [256]
    float* __restrict__ agg) {             // [Ndst, 256]
  __shared__ float Wlds[ED * H];           // 64 KB
  const int tid  = threadIdx.x;
  const int lane = tid & 31;
  const int wave = tid >> 5;

  if (wave == 0) tdm_load_f32(lds_off(&Wlds[0]), W_edge, ED * H);
  __builtin_amdgcn_s_wait_tensorcnt(0);
  __syncthreads();

  const int n_base = wave * 64;
  const int row    = lane & 15;   // A: M index, B/D: N index
  const int khalf  = lane >> 4;
  const int mbase  = khalf * 8;
  const int* __restrict__ srcp = edge_index;
  const int* __restrict__ dstp = edge_index + NE;

  for (int g = 0; g < G_EDGE; ++g) {
    const int e0 = (blockIdx.x * G_EDGE + g) * 16;
    v8f acc[4] = {v8f{}, v8f{}, v8f{}, v8f{}};

    for (int k0 = 0; k0 < ED; k0 += 4) {
      const int ka = k0 + 2 * khalf;
      v2f a = *(const v2f*)(edge_attr + (size_t)(e0 + row) * ED + ka);
#pragma unroll
      for (int t = 0; t < 4; ++t) {
        const int c = n_base + t * 16 + row;
        v2f b;
        b.x = Wlds[ka * H + c];
        b.y = Wlds[ka * H + H + c];
        acc[t] = __builtin_amdgcn_wmma_f32_16x16x4_f32(
            false, a, false, b, (short)0, acc[t], false, false);
      }
    }

#pragma unroll
    for (int j = 0; j < 8; ++j) {
      const int e = e0 + mbase + j;
      const int s = srcp[e];
      const int d = dstp[e];
#pragma unroll
      for (int t = 0; t < 4; ++t) {
        const int c = n_base + t * 16 + row;
        float v = acc[t][j] + b_edge[c] + x_src[(size_t)s * H + c];
        v = v > 0.0f ? v : 0.0f;
        atomicAdd(agg + (size_t)d * H + c, v);
      }
    }
  }
}

// ---------------------------------------------------------------------------
// Node update: out = (agg + (1+eps)*x) @ W_nn + b_nn   [optional relu]
// Block = 256 threads (8 waves); W_nn staged in 32 KB K-chunks via TDM.
// G_NODE row tiles kept live in accumulators across chunks; wave w owns
// 32 cols (2 WMMA col tiles). A operand fused from agg and x on load.
// ---------------------------------------------------------------------------
__global__ __launch_bounds__(256) void gine_node_kernel(
    const float* __restrict__ agg,    // [N, 256]
    const float* __restrict__ x,      // [N, 256]
    const float* __restrict__ W_nn,   // [256, 256]
    const float* __restrict__ b_nn,   // [256]
    const float* __restrict__ eps_p,  // [1]
    float* __restrict__ out,          // [N, 256]
    int do_relu) {
  __shared__ float Wlds[KC * H];      // 32 KB chunk of W_nn
  const int tid   = threadIdx.x;
  const int lane  = tid & 31;
  const int wave  = tid >> 5;
  const int n_base = wave * 32;
  const int row    = lane & 15;
  const int khalf  = lane >> 4;
  const int mbase  = khalf * 8;
  const float ep   = 1.0f + eps_p[0];
  const int r0     = blockIdx.x * (G_NODE * 16);

  v8f acc[G_NODE][2];
#pragma unroll
  for (int g = 0; g < G_NODE; ++g) {
    acc[g][0] = v8f{};
    acc[g][1] = v8f{};
  }

  for (int kc = 0; kc < H; kc += KC) {
    __syncthreads();  // previous chunk fully consumed
    if (wave == 0) tdm_load_f32(lds_off(&Wlds[0]), W_nn + (size_t)kc * H, KC * H);
    __builtin_amdgcn_s_wait_tensorcnt(0);
    __syncthreads();

#pragma unroll
    for (int g = 0; g < G_NODE; ++g) {
      for (int k0 = 0; k0 < KC; k0 += 4) {
        const int kl = k0 + 2 * khalf;
        const size_t off = (size_t)(r0 + g * 16 + row) * H + kc + kl;
        v2f ag = *(const v2f*)(agg + off);
        v2f xd = *(const v2f*)(x + off);
        v2f a;
        a.x = ag.x + ep * xd.x;
        a.y = ag.y + ep * xd.y;
#pragma unroll
        for (int t = 0; t < 2; ++t) {
          const int c = n_base + t * 16 + row;
          v2f b;
          b.x = Wlds[kl * H + c];
          b.y = Wlds[kl * H + H + c];
          acc[g][t] = __builtin_amdgcn_wmma_f32_16x16x4_f32(
              false, a, false, b, (short)0, acc[g][t], false, false);
        }
      }
    }
  }

#pragma unroll
  for (int g = 0; g < G_NODE; ++g) {
#pragma unroll
    for (int j = 0; j < 8; ++j) {
      const int m = r0 + g * 16 + mbase + j;
#pragma unroll
      for (int t = 0; t < 2; ++t) {
        const int c = n_base + t * 16 + row;
        float v = acc[g][t][j] + b_nn[c];
        if (do_relu) v = v > 0.0f ? v : 0.0f;
        out[(size_t)m * H + c] = v;
      }
    }
  }
}

// ---------------------------------------------------------------------------
// Classifier: z = relu(x_book @ W1 + b1) [N,64]; pred = sigmoid(z @ W2 + b2)
// ---------------------------------------------------------------------------
__global__ __launch_bounds__(64) void classifier_kernel(
    const float* __restrict__ xb,   // [N, 256]
    const float* __restrict__ W1,   // [256, 64]
    const float* __restrict__ b1,   // [64]
    const float* __restrict__ W2,   // [64, 1]
    const float* __restrict__ b2,   // [1]
    float* __restrict__ pred) {     // [N]
  const int i = blockIdx.x;
  const int h = threadIdx.x;
  const float* __restrict__ xr = xb + (size_t)i * H;

  float z = b1[h];
  for (int k = 0; k < H; ++k) z = fmaf(xr[k], W1[(size_t)k * 64 + h], z);
  z = z > 0.0f ? z : 0.0f;

  __shared__ float red[64];
  red[h] = z * W2[h];
  __syncthreads();
#pragma unroll
  for (int s = 32; s > 0; s >>= 1) {
    if (h < s) red[h] += red[h + s];
    __syncthreads();
  }
  if (h == 0) pred[i] = 1.0f / (1.0f + expf(-(red[0] + b2[0])));
}

// ---------------------------------------------------------------------------
extern "C" void kernel_launch(void* const* d_in, const int* in_sizes, int n_in,
                              void* d_out, int out_size, void* d_ws,
                              size_t ws_size, hipStream_t stream) {
  const int*   user_nid = (const int*)d_in[0];
  const int*   book_nid = (const int*)d_in[1];
  const int*   ei_u2b   = (const int*)d_in[2];
  const int*   ei_b2u   = (const int*)d_in[3];
  const float* ea_u2b   = (const float*)d_in[4];
  const float* ea_b2u   = (const float*)d_in[5];
  const float* user_emb = (const float*)d_in[6];
  const float* book_emb = (const float*)d_in[7];
  const float* W_edge   = (const float*)d_in[8];
  const float* b_edge   = (const float*)d_in[9];
  const float* W_nn     = (const float*)d_in[10];
  const float* b_nn     = (const float*)d_in[11];
  const float* eps      = (const float*)d_in[12];
  const float* W1       = (const float*)d_in[13];
  const float* b1       = (const float*)d_in[14];
  const float* W2       = (const float*)d_in[15];
  const float* b2       = (const float*)d_in[16];
  float* out = (float*)d_out;

  const size_t slab = (size_t)NU * H;   // 12.8M floats
  float* ws     = (float*)d_ws;
  float* x_user = ws + 0 * slab;
  float* x_book = ws + 1 * slab;
  float* n_user = ws + 2 * slab;
  float* n_book = ws + 3 * slab;
  float* agg_u  = ws + 4 * slab;
  float* agg_b  = ws + 5 * slab;

  float* out_pred = out;
  float* out_user = out + NU;
  float* out_book = out + NU + slab;

  const int edge_blocks = NE / 16 / G_EDGE;   // 3125
  const int node_blocks = NU / 16 / G_NODE;   // 625
  const size_t zn = 2 * slab;
  const int zblocks = (int)((zn + 255) / 256);

  gather_rows_kernel<<<NU, 256, 0, stream>>>(user_emb, user_nid, x_user);
  gather_rows_kernel<<<NB, 256, 0, stream>>>(book_emb, book_nid, x_book);

  // ---------------- layer 0 (relu on outputs) ----------------
  zero_f32_kernel<<<zblocks, 256, 0, stream>>>(agg_u, zn);  // agg_u + agg_b
  gine_edge_kernel<<<edge_blocks, 128, 0, stream>>>(ea_u2b, ei_u2b, x_user,
                                                    W_edge, b_edge, agg_b);
  gine_edge_kernel<<<edge_blocks, 128, 0, stream>>>(ea_b2u, ei_b2u, x_book,
                                                    W_edge, b_edge, agg_u);
  gine_node_kernel<<<node_blocks, 256, 0, stream>>>(agg_b, x_book, W_nn, b_nn,
                                                    eps, n_book, 1);
  gine_node_kernel<<<node_blocks, 256, 0, stream>>>(agg_u, x_user, W_nn, b_nn,
                                                    eps, n_user, 1);

  // ---------------- layer 1 (no relu; write into d_out) ----------------
  zero_f32_kernel<<<zblocks, 256, 0, stream>>>(agg_u, zn);
  gine_edge_kernel<<<edge_blocks, 128, 0, stream>>>(ea_u2b, ei_u2b, n_user,
                                                    W_edge, b_edge, agg_b);
  gine_edge_kernel<<<edge_blocks, 128, 0, stream>>>(ea_b2u, ei_b2u, n_book,
                                                    W_edge, b_edge, agg_u);
  gine_node_kernel<<<node_blocks, 256, 0, stream>>>(agg_b, n_book, W_nn, b_nn,
                                                    eps, out_book, 0);
  gine_node_kernel<<<node_blocks, 256, 0, stream>>>(agg_u, n_user, W_nn, b_nn,
                                                    eps, out_user, 0);

  // ---------------- classifier on x_book ----------------
  classifier_kernel<<<NB, 64, 0, stream>>>(out_book, W1, b1, W2, b2, out_pred);
}